// GAT_72997264163169
// MI455X (gfx1250) — compile-verified
//
#include <hip/hip_runtime.h>
#include <math.h>

#define NODES 50000
#define EDGES 800000
#define FEAT  256     // H * HID = H * OUT
#define NHEAD 4
#define HDIM  64

typedef float v2f __attribute__((ext_vector_type(2)));
typedef float v8f __attribute__((ext_vector_type(8)));

// ---------------------------------------------------------------------------
// float atomic max via sign-split int/uint trick (init to -inf)
// ---------------------------------------------------------------------------
__device__ __forceinline__ void atomic_max_float(float* addr, float val) {
  if (!(val < 0.0f)) {
    atomicMax((int*)addr, __float_as_int(val));
  } else {
    atomicMin((unsigned int*)addr, __float_as_uint(val));
  }
}

__device__ __forceinline__ float leaky_relu02(float v) {
  return v > 0.0f ? v : 0.2f * v;
}

// ---------------------------------------------------------------------------
// GEMM: C[NODES,256] = A[NODES,256] @ B[256,256] using V_WMMA_F32_16X16X4_F32.
// One wave computes a 16x64 strip of C (4 accumulators, A-frag reused 4x).
// 3125 M-tiles x 4 N-strips = 12500 waves. Wave-uniform guard keeps EXEC
// all-ones at WMMA (ISA 7.12 requirement).
// ---------------------------------------------------------------------------
__global__ __launch_bounds__(256) void gat_gemm_wmma(
    const float* __restrict__ A, const float* __restrict__ B,
    float* __restrict__ C) {
  const int lane = threadIdx.x & 31;
  const int wave = blockIdx.x * (blockDim.x >> 5) + (threadIdx.x >> 5);
  const int MTILES = NODES / 16;          // 3125 exactly
  if (wave >= MTILES * 4) return;         // wave-uniform
  const int mtile = wave >> 2;
  const int nb0   = (wave & 3) * 64;

  const int lh = lane & 15;               // row/col within tile
  const int kh = (lane >> 4) << 1;        // K sub-offset: 0 or 2

  const float* Arow = A + (size_t)(mtile * 16 + lh) * FEAT;

  v8f c0 = {}, c1 = {}, c2 = {}, c3 = {};
  for (int k = 0; k < FEAT; k += 4) {
    v2f a;
    a.x = Arow[k + kh];
    a.y = Arow[k + kh + 1];
    const float* Bk0 = B + (size_t)(k + kh) * FEAT + nb0 + lh;
    const float* Bk1 = Bk0 + FEAT;
    v2f b0; b0.x = Bk0[0];  b0.y = Bk1[0];
    v2f b1; b1.x = Bk0[16]; b1.y = Bk1[16];
    v2f b2; b2.x = Bk0[32]; b2.y = Bk1[32];
    v2f b3; b3.x = Bk0[48]; b3.y = Bk1[48];
    c0 = __builtin_amdgcn_wmma_f32_16x16x4_f32(false, a, false, b0, (short)0, c0, false, false);
    c1 = __builtin_amdgcn_wmma_f32_16x16x4_f32(false, a, false, b1, (short)0, c1, false, false);
    c2 = __builtin_amdgcn_wmma_f32_16x16x4_f32(false, a, false, b2, (short)0, c2, false, false);
    c3 = __builtin_amdgcn_wmma_f32_16x16x4_f32(false, a, false, b3, (short)0, c3, false, false);
  }

  // C/D layout: VGPR i holds M = i + (lane>=16 ? 8 : 0), N = lane&15
  const int crow = mtile * 16 + ((lane >> 4) << 3);
  float* Crow = C + (size_t)crow * FEAT + nb0 + lh;
#pragma unroll
  for (int i = 0; i < 8; ++i) {
    Crow[(size_t)i * FEAT + 0]  = c0[i];
    Crow[(size_t)i * FEAT + 16] = c1[i];
    Crow[(size_t)i * FEAT + 32] = c2[i];
    Crow[(size_t)i * FEAT + 48] = c3[i];
  }
}

// ---------------------------------------------------------------------------
// el[n,h] = dot(feat[n,h,:], al[h,:]); er likewise; init m=-inf, den=0
// ---------------------------------------------------------------------------
__global__ __launch_bounds__(256) void gat_elr(
    const float* __restrict__ feat, const float* __restrict__ al,
    const float* __restrict__ ar, float* __restrict__ el,
    float* __restrict__ er, float* __restrict__ m, float* __restrict__ den) {
  const int t = blockIdx.x * blockDim.x + threadIdx.x;
  if (t >= NODES * NHEAD) return;
  const int n = t >> 2, h = t & 3;
  const float* f = feat + (size_t)n * FEAT + h * HDIM;
  const float* a = al + h * HDIM;
  const float* r = ar + h * HDIM;
  float sl = 0.0f, sr = 0.0f;
#pragma unroll 8
  for (int d = 0; d < HDIM; ++d) {
    const float v = f[d];
    sl += v * a[d];
    sr += v * r[d];
  }
  el[t]  = sl;
  er[t]  = sr;
  m[t]   = -INFINITY;
  den[t] = 0.0f;
}

// ---------------------------------------------------------------------------
// rst init: residual + bias (segment sums atomically accumulated on top)
// ---------------------------------------------------------------------------
__global__ __launch_bounds__(256) void gat_rst_init(
    const float* __restrict__ hin, const float* __restrict__ bias,
    float* __restrict__ rst) {
  const int t = blockIdx.x * blockDim.x + threadIdx.x;
  if (t >= NODES * FEAT) return;
  rst[t] = hin[t] + bias[t & (FEAT - 1)];
}

// ---------------------------------------------------------------------------
// edge pass 1: e = leaky_relu(el[src]+er[dst]); atomic segment max into m[dst]
// ---------------------------------------------------------------------------
__global__ __launch_bounds__(256) void gat_edge_max(
    const int* __restrict__ src, const int* __restrict__ dst,
    const float* __restrict__ el, const float* __restrict__ er,
    float* __restrict__ ebuf, float* __restrict__ m) {
  const int e = blockIdx.x * blockDim.x + threadIdx.x;
  if (e >= EDGES) return;
  const int s = src[e], d = dst[e];
  const float4 l = *(const float4*)(el + (size_t)s * NHEAD);
  const float4 r = *(const float4*)(er + (size_t)d * NHEAD);
  float4 v;
  v.x = leaky_relu02(l.x + r.x);
  v.y = leaky_relu02(l.y + r.y);
  v.z = leaky_relu02(l.z + r.z);
  v.w = leaky_relu02(l.w + r.w);
  *(float4*)(ebuf + (size_t)e * NHEAD) = v;
  float* md = m + (size_t)d * NHEAD;
  atomic_max_float(md + 0, v.x);
  atomic_max_float(md + 1, v.y);
  atomic_max_float(md + 2, v.z);
  atomic_max_float(md + 3, v.w);
}

// ---------------------------------------------------------------------------
// edge pass 2: ex = exp(e - m[dst]); atomic segment sum into den[dst]
// ---------------------------------------------------------------------------
__global__ __launch_bounds__(256) void gat_edge_exp(
    const int* __restrict__ dst, float* __restrict__ ebuf,
    const float* __restrict__ m, float* __restrict__ den) {
  const int e = blockIdx.x * blockDim.x + threadIdx.x;
  if (e >= EDGES) return;
  const int d = dst[e];
  const float4 v  = *(const float4*)(ebuf + (size_t)e * NHEAD);
  const float4 mm = *(const float4*)(m + (size_t)d * NHEAD);
  float4 x;
  x.x = expf(v.x - mm.x);
  x.y = expf(v.y - mm.y);
  x.z = expf(v.z - mm.z);
  x.w = expf(v.w - mm.w);
  *(float4*)(ebuf + (size_t)e * NHEAD) = x;
  float* dd = den + (size_t)d * NHEAD;
  atomicAdd(dd + 0, x.x);
  atomicAdd(dd + 1, x.y);
  atomicAdd(dd + 2, x.z);
  atomicAdd(dd + 3, x.w);
}

// ---------------------------------------------------------------------------
// edge pass 3: rst[dst] += (ex/den[dst]) * feat[src]  — one wave per edge,
// lane i handles feat elements [i*8, i*8+8) (head = i>>3). L2-resident.
// ---------------------------------------------------------------------------
__global__ __launch_bounds__(256) void gat_edge_aggr(
    const int* __restrict__ src, const int* __restrict__ dst,
    const float* __restrict__ ebuf, const float* __restrict__ den,
    const float* __restrict__ feat, float* __restrict__ rst) {
  const int lane = threadIdx.x & 31;
  const int e = blockIdx.x * (blockDim.x >> 5) + (threadIdx.x >> 5);
  if (e >= EDGES) return;
  const int s = src[e], d = dst[e];
  const float4 x  = *(const float4*)(ebuf + (size_t)e * NHEAD);
  const float4 dn = *(const float4*)(den + (size_t)d * NHEAD);
  const int hsel = lane >> 3;
  const float ex = (hsel == 0) ? x.x : (hsel == 1) ? x.y : (hsel == 2) ? x.z : x.w;
  const float dd = (hsel == 0) ? dn.x : (hsel == 1) ? dn.y : (hsel == 2) ? dn.z : dn.w;
  const float alpha = ex / dd;
  const float* fs = feat + (size_t)s * FEAT + lane * 8;
  float* rd       = rst  + (size_t)d * FEAT + lane * 8;
  const float4 f0 = *(const float4*)(fs);
  const float4 f1 = *(const float4*)(fs + 4);
  atomicAdd(rd + 0, f0.x * alpha);
  atomicAdd(rd + 1, f0.y * alpha);
  atomicAdd(rd + 2, f0.z * alpha);
  atomicAdd(rd + 3, f0.w * alpha);
  atomicAdd(rd + 4, f1.x * alpha);
  atomicAdd(rd + 5, f1.y * alpha);
  atomicAdd(rd + 6, f1.z * alpha);
  atomicAdd(rd + 7, f1.w * alpha);
}

// ---------------------------------------------------------------------------
// final: out[n,d] = mean_h relu(rst[n,h,d])
// ---------------------------------------------------------------------------
__global__ __launch_bounds__(256) void gat_finalize(
    const float* __restrict__ rst, float* __restrict__ out) {
  const int t = blockIdx.x * blockDim.x + threadIdx.x;
  if (t >= NODES * HDIM) return;
  const int n = t >> 6, dcol = t & (HDIM - 1);
  const float* r = rst + (size_t)n * FEAT + dcol;
  float s = 0.0f;
#pragma unroll
  for (int h = 0; h < NHEAD; ++h) s += fmaxf(r[h * HDIM], 0.0f);
  out[t] = 0.25f * s;
}

// ---------------------------------------------------------------------------
extern "C" void kernel_launch(void* const* d_in, const int* in_sizes, int n_in,
                              void* d_out, int out_size, void* d_ws, size_t ws_size,
                              hipStream_t stream) {
  (void)in_sizes; (void)n_in; (void)out_size; (void)ws_size;

  const float* features = (const float*)d_in[0];
  const int*   src      = (const int*)d_in[1];
  const int*   dst      = (const int*)d_in[2];
  const float* W[3]  = {(const float*)d_in[3], (const float*)d_in[7],  (const float*)d_in[11]};
  const float* al[3] = {(const float*)d_in[4], (const float*)d_in[8],  (const float*)d_in[12]};
  const float* ar[3] = {(const float*)d_in[5], (const float*)d_in[9],  (const float*)d_in[13]};
  const float* bb[3] = {(const float*)d_in[6], (const float*)d_in[10], (const float*)d_in[14]};

  float* ws = (float*)d_ws;
  size_t off = 0;
  float* featbuf = ws + off; off += (size_t)NODES * FEAT;   // 12.8M
  float* rstA    = ws + off; off += (size_t)NODES * FEAT;   // 12.8M
  float* rstB    = ws + off; off += (size_t)NODES * FEAT;   // 12.8M
  float* el      = ws + off; off += (size_t)NODES * NHEAD;
  float* er      = ws + off; off += (size_t)NODES * NHEAD;
  float* mbuf    = ws + off; off += (size_t)NODES * NHEAD;
  float* den     = ws + off; off += (size_t)NODES * NHEAD;
  float* ebuf    = ws + off; off += (size_t)EDGES * NHEAD;

  const float* hin[3] = {features, rstA, rstB};
  float*      hout[3] = {rstA,     rstB, rstA};

  const int gemm_waves  = (NODES / 16) * 4;                 // 12500
  const int gemm_blocks = (gemm_waves + 7) / 8;             // 1563
  const int elr_blocks  = (NODES * NHEAD + 255) / 256;
  const int rst_blocks  = (NODES * FEAT + 255) / 256;
  const int edge_blocks = (EDGES + 255) / 256;
  const int aggr_blocks = (EDGES + 7) / 8;                  // one wave per edge
  const int fin_blocks  = (NODES * HDIM + 255) / 256;

  for (int L = 0; L < 3; ++L) {
    gat_gemm_wmma<<<gemm_blocks, 256, 0, stream>>>(hin[L], W[L], featbuf);
    gat_elr<<<elr_blocks, 256, 0, stream>>>(featbuf, al[L], ar[L], el, er, mbuf, den);
    gat_rst_init<<<rst_blocks, 256, 0, stream>>>(hin[L], bb[L], hout[L]);
    gat_edge_max<<<edge_blocks, 256, 0, stream>>>(src, dst, el, er, ebuf, mbuf);
    gat_edge_exp<<<edge_blocks, 256, 0, stream>>>(dst, ebuf, mbuf, den);
    gat_edge_aggr<<<aggr_blocks, 256, 0, stream>>>(src, dst, ebuf, den, featbuf, hout[L]);
  }
  gat_finalize<<<fin_blocks, 256, 0, stream>>>(rstA, (float*)d_out);
}